// RelativeGridAttention_74491912782186
// MI455X (gfx1250) — compile-verified
//
#include <hip/hip_runtime.h>

typedef float v2f __attribute__((ext_vector_type(2)));
typedef float v8f __attribute__((ext_vector_type(8)));

#define B_  8
#define Q_  256
#define C_  64
#define HW_ 4096      // 64*64
#define RG_ 16129     // 127*127
#define RW_ 127

__device__ __forceinline__ v8f wmma_f32_k4(v2f a, v2f b, v8f c) {
    return __builtin_amdgcn_wmma_f32_16x16x4_f32(
        /*neg_a=*/false, a, /*neg_b=*/false, b,
        /*c_mod=*/(short)0, c, /*reuse_a=*/false, /*reuse_b=*/false);
}

// ---------------------------------------------------------------------------
// qb[q, p] = sum_c queries[b,q,c] * rel_bias[p, c]   (p = u*127+v, full grid)
// grid: (16 m-tiles, 253 n-groups), block 128 (4 waves, 1 n-tile each)
// ---------------------------------------------------------------------------
__global__ void rga_qb_kernel(const float* __restrict__ q,
                              const float* __restrict__ rb,
                              float* __restrict__ qb, int b) {
    const int tid  = threadIdx.x;
    const int wave = tid >> 5;
    const int lane = tid & 31;
    const int hl   = lane >> 4;   // lane half
    const int ln   = lane & 15;   // column / row-in-half

    const int nt = blockIdx.y * 4 + wave;     // n-tile id, 0..1008
    if (nt >= 1009) return;                   // uniform per wave -> EXEC stays full
    const int m0 = blockIdx.x * 16;
    const int p0 = nt * 16;

    const float* __restrict__ Qb = q + ((size_t)b * Q_ + m0) * C_;

    int p = p0 + ln;
    int pc = p > (RG_ - 1) ? (RG_ - 1) : p;   // clamp tail loads (cols not stored)

    v8f c = {0.f, 0.f, 0.f, 0.f, 0.f, 0.f, 0.f, 0.f};
#pragma unroll
    for (int ks = 0; ks < 16; ++ks) {
        const int k = ks * 4 + hl * 2;
        v2f av, bv;
        av[0] = Qb[ln * C_ + k];
        av[1] = Qb[ln * C_ + k + 1];
        bv[0] = rb[(size_t)pc * C_ + k];
        bv[1] = rb[(size_t)pc * C_ + k + 1];
        c = wmma_f32_k4(av, bv, c);
    }

    if (p < RG_) {
#pragma unroll
        for (int r = 0; r < 8; ++r) {
            const int m = hl * 8 + r;
            qb[(size_t)(m0 + m) * RG_ + p] = c[r];
        }
    }
}

// ---------------------------------------------------------------------------
// scores[b,q,hw] = sum_c Q[b,q,c]*K[b,hw,c]  +  qb[q, window(hw)]
// grid: (16 m-tiles, 8 n-chunks), block 256 (8 waves, 4 n-tiles each)
// ---------------------------------------------------------------------------
__global__ void rga_scores_kernel(const float* __restrict__ q,
                                  const float* __restrict__ keys,
                                  const int* __restrict__ pos,
                                  const float* __restrict__ qb,
                                  float* __restrict__ scores, int b) {
    const int tid  = threadIdx.x;
    const int wave = tid >> 5;
    const int lane = tid & 31;
    const int hl   = lane >> 4;
    const int ln   = lane & 15;
    const int m0   = blockIdx.x * 16;

    __shared__ int qboff[16];
    if (tid < 16) {
        const int qi = m0 + tid;
        const int ph = pos[((size_t)b * Q_ + qi) * 2 + 0];
        const int pw = pos[((size_t)b * Q_ + qi) * 2 + 1];
        qboff[tid] = qi * RG_ + (63 - ph) * RW_ + (63 - pw);
    }
    __syncthreads();

    // Preload this wave's A fragments (16 queries x 64 K) once; reuse for all tiles.
    const float* __restrict__ Qb = q + ((size_t)b * Q_ + m0) * C_;
    float a0[16], a1[16];
#pragma unroll
    for (int ks = 0; ks < 16; ++ks) {
        const int k = ks * 4 + hl * 2;
        a0[ks] = Qb[ln * C_ + k];
        a1[ks] = Qb[ln * C_ + k + 1];
    }

    const float* __restrict__ Kb = keys + (size_t)b * HW_ * C_;

#pragma unroll 1
    for (int t = 0; t < 4; ++t) {
        const int nt  = blockIdx.y * 32 + wave * 4 + t;  // 0..255
        const int hw0 = nt * 16;
        const int hrow = hw0 >> 6;
        const int w0   = hw0 & 63;

        v8f c = {0.f, 0.f, 0.f, 0.f, 0.f, 0.f, 0.f, 0.f};
#pragma unroll
        for (int ks = 0; ks < 16; ++ks) {
            const int k = ks * 4 + hl * 2;
            v2f av, bv;
            av[0] = a0[ks];
            av[1] = a1[ks];
            bv[0] = Kb[(size_t)(hw0 + ln) * C_ + k];
            bv[1] = Kb[(size_t)(hw0 + ln) * C_ + k + 1];
            c = wmma_f32_k4(av, bv, c);
        }

#pragma unroll
        for (int r = 0; r < 8; ++r) {
            const int m    = hl * 8 + r;
            const float bv = qb[(size_t)qboff[m] + hrow * RW_ + w0 + ln];
            scores[((size_t)(b * Q_ + m0 + m)) * HW_ + hw0 + ln] = c[r] + bv;
        }
    }
}

// ---------------------------------------------------------------------------
// In-place softmax over rows of 4096. grid 2048, block 256, 16 vals/thread.
// ---------------------------------------------------------------------------
__global__ void rga_softmax_kernel(float* __restrict__ scores) {
    const int tid = threadIdx.x;
    float* __restrict__ row = scores + (size_t)blockIdx.x * HW_;

    float v[16];
#pragma unroll
    for (int i = 0; i < 16; ++i) v[i] = row[i * 256 + tid];

    float mx = v[0];
#pragma unroll
    for (int i = 1; i < 16; ++i) mx = fmaxf(mx, v[i]);

    __shared__ float red[256];
    red[tid] = mx;
    __syncthreads();
    for (int s = 128; s > 0; s >>= 1) {
        if (tid < s) red[tid] = fmaxf(red[tid], red[tid + s]);
        __syncthreads();
    }
    mx = red[0];
    __syncthreads();

    float sum = 0.f;
#pragma unroll
    for (int i = 0; i < 16; ++i) {
        v[i] = __expf(v[i] - mx);
        sum += v[i];
    }
    red[tid] = sum;
    __syncthreads();
    for (int s = 128; s > 0; s >>= 1) {
        if (tid < s) red[tid] += red[tid + s];
        __syncthreads();
    }
    const float inv = 1.0f / red[0];

#pragma unroll
    for (int i = 0; i < 16; ++i) row[i * 256 + tid] = v[i] * inv;
}

// ---------------------------------------------------------------------------
// out = y + conv2(relu(conv1(y)))  with 1-channel 3x3 convs, SAME zero pad.
// grid 2048 (one 64x64 map), block 256, 32KB LDS.
// ---------------------------------------------------------------------------
__global__ void rga_conv_kernel(const float* __restrict__ attn,
                                const float* __restrict__ w1,
                                const float* __restrict__ b1,
                                const float* __restrict__ w2,
                                const float* __restrict__ b2,
                                float* __restrict__ out) {
    __shared__ float t0[HW_];
    __shared__ float t1[HW_];
    const int tid = threadIdx.x;
    const size_t base = (size_t)blockIdx.x * HW_;

    float cw1[9], cw2[9];
#pragma unroll
    for (int k = 0; k < 9; ++k) { cw1[k] = w1[k]; cw2[k] = w2[k]; }
    const float fb1 = b1[0], fb2 = b2[0];

#pragma unroll
    for (int i = 0; i < 16; ++i) {
        const int j = i * 256 + tid;
        t0[j] = attn[base + j];
    }
    __syncthreads();

#pragma unroll 1
    for (int i = 0; i < 16; ++i) {
        const int j = i * 256 + tid;
        const int h = j >> 6, w = j & 63;
        float acc = fb1;
#pragma unroll
        for (int dy = 0; dy < 3; ++dy)
#pragma unroll
            for (int dx = 0; dx < 3; ++dx) {
                const int hh = h + dy - 1, ww = w + dx - 1;
                if (hh >= 0 && hh < 64 && ww >= 0 && ww < 64)
                    acc += cw1[dy * 3 + dx] * t0[hh * 64 + ww];
            }
        t1[j] = fmaxf(acc, 0.f);
    }
    __syncthreads();

#pragma unroll 1
    for (int i = 0; i < 16; ++i) {
        const int j = i * 256 + tid;
        const int h = j >> 6, w = j & 63;
        float acc = fb2;
#pragma unroll
        for (int dy = 0; dy < 3; ++dy)
#pragma unroll
            for (int dx = 0; dx < 3; ++dx) {
                const int hh = h + dy - 1, ww = w + dx - 1;
                if (hh >= 0 && hh < 64 && ww >= 0 && ww < 64)
                    acc += cw2[dy * 3 + dx] * t1[hh * 64 + ww];
            }
        out[base + j] = t0[j] + acc;
    }
}

// ---------------------------------------------------------------------------
extern "C" void kernel_launch(void* const* d_in, const int* in_sizes, int n_in,
                              void* d_out, int out_size, void* d_ws, size_t ws_size,
                              hipStream_t stream) {
    const float* queries = (const float*)d_in[0];
    const float* keys    = (const float*)d_in[1];
    const int*   pos     = (const int*)d_in[2];
    const float* rel     = (const float*)d_in[3];
    const float* c1w     = (const float*)d_in[4];
    const float* c1b     = (const float*)d_in[5];
    const float* c2w     = (const float*)d_in[6];
    const float* c2b     = (const float*)d_in[7];
    float* outp = (float*)d_out;

    // ws layout: [qb: 256*16129 f32 = 16,516,096 B][scores: 8*256*4096 f32 = 33,554,432 B]
    char* wsb = (char*)d_ws;
    float* qb_ws     = (float*)wsb;
    float* scores_ws = (float*)(wsb + (size_t)Q_ * RG_ * sizeof(float));

    for (int b = 0; b < B_; ++b) {
        rga_qb_kernel<<<dim3(16, 253), 128, 0, stream>>>(queries, rel, qb_ws, b);
        rga_scores_kernel<<<dim3(16, 8), 256, 0, stream>>>(queries, keys, pos,
                                                           qb_ws, scores_ws, b);
    }
    rga_softmax_kernel<<<B_ * Q_, 256, 0, stream>>>(scores_ws);
    rga_conv_kernel<<<B_ * Q_, 256, 0, stream>>>(scores_ws, c1w, c1b, c2w, c2b, outp);
}